// SteadyStateGMRES_20212116095478
// MI455X (gfx1250) — compile-verified
//
#include <hip/hip_runtime.h>
#include <math.h>

// ---------------------------------------------------------------------------
// Steady-state Lindblad GMRES on MI455X (gfx1250, wave32).
// All 128x128 complex GEMMs run on V_WMMA_F32_16X16X4_F32. GEMM operands are
// staged through LDS (coalesced b128 global loads, double-buffered panels,
// transpose/conj folded into the staging) so the WMMA pipe is fed from
// ds_load instead of lane-strided global loads. The eig-based Lyapunov
// preconditioner is replaced by a Richardson iteration on G X + X G^dag = Y
// (pure GEMMs -> WMMA pipes). All reduction scalars live in device workspace
// so the fixed-length GMRES is fully graph-capturable.
// ---------------------------------------------------------------------------

#define NH     128
#define NN     (NH * NH)          // 16384 complex elements per matrix/vector
#define CV     (2 * NN)           // floats per complex vector (re plane, im plane)
#define KRY    32
#define MAXIT  4
#define LYAP_IT 4
#define EPSV   1e-20f
#define TOLV   1e-4f

#define BK     32                 // K-panel depth
#define LP     36                 // padded LDS row stride (16B-aligned, bank-safe)

typedef __attribute__((ext_vector_type(2))) float v2f;
typedef __attribute__((ext_vector_type(8))) float v8f;

// ---------------------------------------------------------------------------
// Stage one 32xBK A panel and one BKx32 B panel into LDS (canonical layout
// As[row][k], Bs[k][col]); transpose + conjugation folded in for op==1.
// 128 threads; float4 (b128) global loads, coalesced in both op modes.
// ---------------------------------------------------------------------------
__device__ __forceinline__ void stage_panels(
    float (*__restrict__ Asr)[LP], float (*__restrict__ Asi)[LP],
    float (*__restrict__ Bsr)[LP], float (*__restrict__ Bsi)[LP],
    const float* __restrict__ Ar, const float* __restrict__ Ai, int opA,
    const float* __restrict__ Br, const float* __restrict__ Bi, int opB,
    int bm, int bn, int kb, int tid)
{
    if (opA == 0) {
        for (int idx = tid; idx < 256; idx += 128) {
            const int row = idx >> 3;
            const int kq  = (idx & 7) << 2;
            const float4 vr = *(const float4*)(Ar + (bm + row) * NH + kb + kq);
            const float4 vi = *(const float4*)(Ai + (bm + row) * NH + kb + kq);
            Asr[row][kq + 0] = vr.x; Asr[row][kq + 1] = vr.y;
            Asr[row][kq + 2] = vr.z; Asr[row][kq + 3] = vr.w;
            Asi[row][kq + 0] = vi.x; Asi[row][kq + 1] = vi.y;
            Asi[row][kq + 2] = vi.z; Asi[row][kq + 3] = vi.w;
        }
    } else {  // As[row][k] = conj(A[kb+k][bm+row])
        for (int idx = tid; idx < 256; idx += 128) {
            const int k  = idx >> 3;
            const int rq = (idx & 7) << 2;
            const float4 vr = *(const float4*)(Ar + (kb + k) * NH + bm + rq);
            const float4 vi = *(const float4*)(Ai + (kb + k) * NH + bm + rq);
            Asr[rq + 0][k] = vr.x; Asr[rq + 1][k] = vr.y;
            Asr[rq + 2][k] = vr.z; Asr[rq + 3][k] = vr.w;
            Asi[rq + 0][k] = -vi.x; Asi[rq + 1][k] = -vi.y;
            Asi[rq + 2][k] = -vi.z; Asi[rq + 3][k] = -vi.w;
        }
    }
    if (opB == 0) {
        for (int idx = tid; idx < 256; idx += 128) {
            const int k  = idx >> 3;
            const int cq = (idx & 7) << 2;
            const float4 vr = *(const float4*)(Br + (kb + k) * NH + bn + cq);
            const float4 vi = *(const float4*)(Bi + (kb + k) * NH + bn + cq);
            Bsr[k][cq + 0] = vr.x; Bsr[k][cq + 1] = vr.y;
            Bsr[k][cq + 2] = vr.z; Bsr[k][cq + 3] = vr.w;
            Bsi[k][cq + 0] = vi.x; Bsi[k][cq + 1] = vi.y;
            Bsi[k][cq + 2] = vi.z; Bsi[k][cq + 3] = vi.w;
        }
    } else {  // Bs[k][col] = conj(B[bn+col][kb+k])
        for (int idx = tid; idx < 256; idx += 128) {
            const int col = idx >> 3;
            const int kq  = (idx & 7) << 2;
            const float4 vr = *(const float4*)(Br + (bn + col) * NH + kb + kq);
            const float4 vi = *(const float4*)(Bi + (bn + col) * NH + kb + kq);
            Bsr[kq + 0][col] = vr.x; Bsr[kq + 1][col] = vr.y;
            Bsr[kq + 2][col] = vr.z; Bsr[kq + 3][col] = vr.w;
            Bsi[kq + 0][col] = -vi.x; Bsi[kq + 1][col] = -vi.y;
            Bsi[kq + 2][col] = -vi.z; Bsi[kq + 3][col] = -vi.w;
        }
    }
}

// ---------------------------------------------------------------------------
// Complex GEMM: C = alpha * op(A) op(B) + beta * C, 128x128, split re/im.
// op: 0 = N, 1 = conjugate-transpose (dagger).
// 4 waves/block, each owns a 16x16 tile of a 32x32 macro-tile; 16 blocks.
// Double-buffered LDS panels; WMMA fed from ds_load.
// ---------------------------------------------------------------------------
__global__ __launch_bounds__(128)
void czgemm128(const float* __restrict__ Ar, const float* __restrict__ Ai, int opA,
               const float* __restrict__ Br, const float* __restrict__ Bi, int opB,
               float* __restrict__ Cr, float* __restrict__ Ci,
               float alr, float ali, float beta)
{
    __shared__ float Asr[2][32][LP], Asi[2][32][LP];
    __shared__ float Bsr[2][32][LP], Bsi[2][32][LP];

    const int tid  = threadIdx.x;
    const int wave = tid >> 5;
    const int lane = tid & 31;
    const int bm   = (blockIdx.x >> 2) << 5;
    const int bn   = (blockIdx.x & 3) << 5;
    const int aoff = (wave >> 1) << 4;          // sub-tile row origin in macro-tile
    const int boff = (wave & 1) << 4;           // sub-tile col origin
    const int m    = lane & 15;
    const int kg   = (lane >> 4) << 1;          // 0 or 2 (K group per half-wave)

    v8f acc_rr = {}, acc_ii = {}, acc_ri = {}, acc_ir = {};

    stage_panels(Asr[0], Asi[0], Bsr[0], Bsi[0], Ar, Ai, opA, Br, Bi, opB,
                 bm, bn, 0, tid);
    __syncthreads();

    for (int kb = 0; kb < NH; kb += BK) {
        const int cur = (kb >> 5) & 1;
        if (kb + BK < NH) {
            stage_panels(Asr[1 - cur], Asi[1 - cur], Bsr[1 - cur], Bsi[1 - cur],
                         Ar, Ai, opA, Br, Bi, opB, bm, bn, kb + BK, tid);
        }
        if (kb + 2 * BK < NH) {                 // warm L2/WGP$ for the K+2 panels
            __builtin_prefetch(Ar + (bm + (tid & 31)) * NH + kb + 2 * BK, 0, 1);
            __builtin_prefetch(Br + (kb + 2 * BK + (tid & 31)) * NH + bn, 0, 1);
        }

        const float (*__restrict__ pAr)[LP] = Asr[cur];
        const float (*__restrict__ pAi)[LP] = Asi[cur];
        const float (*__restrict__ pBr)[LP] = Bsr[cur];
        const float (*__restrict__ pBi)[LP] = Bsi[cur];

#pragma unroll
        for (int k4 = 0; k4 < BK; k4 += 4) {
            v2f ar, ai, br, bi;
#pragma unroll
            for (int r = 0; r < 2; ++r) {
                const int kk = k4 + kg + r;
                ar[r] = pAr[aoff + m][kk];
                ai[r] = pAi[aoff + m][kk];
                br[r] = pBr[kk][boff + m];
                bi[r] = pBi[kk][boff + m];
            }
            acc_rr = __builtin_amdgcn_wmma_f32_16x16x4_f32(false, ar, false, br, (short)0, acc_rr, false, false);
            acc_ii = __builtin_amdgcn_wmma_f32_16x16x4_f32(false, ai, false, bi, (short)0, acc_ii, false, false);
            acc_ri = __builtin_amdgcn_wmma_f32_16x16x4_f32(false, ar, false, bi, (short)0, acc_ri, false, false);
            acc_ir = __builtin_amdgcn_wmma_f32_16x16x4_f32(false, ai, false, br, (short)0, acc_ir, false, false);
        }
        __syncthreads();
    }

    const int row0 = bm + aoff + ((lane >> 4) << 3);   // lanes 16-31 hold M = v + 8
    const int col  = bn + boff + (lane & 15);
#pragma unroll
    for (int v = 0; v < 8; ++v) {
        const int idx = (row0 + v) * NH + col;
        const float pr = acc_rr[v] - acc_ii[v];
        const float pi = acc_ri[v] + acc_ir[v];
        float outr = alr * pr - ali * pi;
        float outi = alr * pi + ali * pr;
        if (beta != 0.f) { outr += beta * Cr[idx]; outi += beta * Ci[idx]; }
        Cr[idx] = outr;
        Ci[idx] = outi;
    }
}

// ---------------------------------------------------------------------------
// Elementwise / small kernels. cvec layout: [re(NN) | im(NN)].
// Elementwise launches: <<<64,256>>> covering NN exactly.
// ---------------------------------------------------------------------------
__global__ __launch_bounds__(256)
void k_copy(float* __restrict__ dst, const float* __restrict__ src) {
    const int e = blockIdx.x * 256 + threadIdx.x;
    dst[e] = src[e];
    dst[NN + e] = src[NN + e];
}

__global__ __launch_bounds__(256)
void k_scale_host(float* __restrict__ x, float sr, float si) {
    const int e = blockIdx.x * 256 + threadIdx.x;
    const float xr = x[e], xi = x[NN + e];
    x[e]      = sr * xr - si * xi;
    x[NN + e] = sr * xi + si * xr;
}

__global__ __launch_bounds__(256)
void k_scale_dev(float* __restrict__ dst, const float* __restrict__ src,
                 const float* __restrict__ tauSlot) {
    const float t = tauSlot[0];
    const int e = blockIdx.x * 256 + threadIdx.x;
    dst[e]      = t * src[e];
    dst[NN + e] = t * src[NN + e];
}

__global__ __launch_bounds__(256)
void k_axpy_dev(float* __restrict__ x, const float* __restrict__ r,
                const float* __restrict__ tauSlot) {
    const float t = tauSlot[0];
    const int e = blockIdx.x * 256 + threadIdx.x;
    x[e]      += t * r[e];
    x[NN + e] += t * r[NN + e];
}

__global__ __launch_bounds__(256)
void k_buildG(float* __restrict__ G, const float* __restrict__ Hr,
              const float* __restrict__ Hi, const float* __restrict__ LdagL) {
    const int e = blockIdx.x * 256 + threadIdx.x;
    G[e]      = -Hi[e] + 0.5f * LdagL[e];       // Re(iH + 0.5 LdagL)
    G[NN + e] =  Hr[e] + 0.5f * LdagL[NN + e];  // Im
}

__global__ __launch_bounds__(256)
void k_set_identity(float* __restrict__ x) {
    const int e = blockIdx.x * 256 + threadIdx.x;
    x[e]      = ((e >> 7) == (e & 127)) ? 1.f : 0.f;
    x[NN + e] = 0.f;
}

__global__ __launch_bounds__(1)
void k_set_entry(float* __restrict__ x) { x[0] = 1.f; }

__global__ __launch_bounds__(256)
void k_neg_plus_id(float* __restrict__ x) {    // x = vec(I) - x
    const int e = blockIdx.x * 256 + threadIdx.x;
    const float d = ((e >> 7) == (e & 127)) ? 1.f : 0.f;
    x[e]      = d - x[e];
    x[NN + e] = -x[NN + e];
}

__global__ __launch_bounds__(128)
void k_add_diag(float* __restrict__ out, const float* __restrict__ tr) {
    const int d = threadIdx.x * (NH + 1);
    out[d]      += tr[0];
    out[NN + d] += tr[1];
}

// trace reduction over the diagonal (u = vec(I) dots are traces)
__global__ __launch_bounds__(128)
void k_trace(const float* __restrict__ x, float* __restrict__ dst, int addone) {
    __shared__ float sr[128], si[128];
    const int i = threadIdx.x;
    const int d = i * (NH + 1);
    sr[i] = x[d];
    si[i] = x[NN + d];
    __syncthreads();
    for (int o = 64; o > 0; o >>= 1) {
        if (i < o) { sr[i] += sr[i + o]; si[i] += si[i + o]; }
        __syncthreads();
    }
    if (i == 0) {
        dst[0] = (addone ? 1.f : 0.f) + sr[0];
        dst[1] = si[0];
    }
}

__global__ __launch_bounds__(256)
void k_norm(const float* __restrict__ v, float* __restrict__ dst) {
    __shared__ float sh[256];
    float s = 0.f;
    for (int e = threadIdx.x; e < NN; e += 256) {
        const float r = v[e], i = v[NN + e];
        s += r * r + i * i;
    }
    sh[threadIdx.x] = s;
    __syncthreads();
    for (int o = 128; o > 0; o >>= 1) {
        if (threadIdx.x < o) sh[threadIdx.x] += sh[threadIdx.x + o];
        __syncthreads();
    }
    if (threadIdx.x == 0) dst[0] = sqrtf(sh[0]);
}

__global__ __launch_bounds__(256)
void k_fro_tau(const float* __restrict__ g, float* __restrict__ dst) {
    __shared__ float sh[256];
    float s = 0.f;
    for (int e = threadIdx.x; e < NN; e += 256) {
        const float r = g[e], i = g[NN + e];
        s += r * r + i * i;
    }
    sh[threadIdx.x] = s;
    __syncthreads();
    for (int o = 128; o > 0; o >>= 1) {
        if (threadIdx.x < o) sh[threadIdx.x] += sh[threadIdx.x + o];
        __syncthreads();
    }
    if (threadIdx.x == 0) dst[0] = 0.5f / (sqrtf(sh[0]) + 1e-6f);  // Richardson tau
}

__global__ __launch_bounds__(256)
void k_maxabs(const float* __restrict__ v, float* __restrict__ dst) {
    __shared__ float sh[256];
    float s = 0.f;
    for (int e = threadIdx.x; e < NN; e += 256) {
        const float r = v[e], i = v[NN + e];
        const float a = sqrtf(r * r + i * i);
        s = fmaxf(s, a);
    }
    sh[threadIdx.x] = s;
    __syncthreads();
    for (int o = 128; o > 0; o >>= 1) {
        if (threadIdx.x < o) sh[threadIdx.x] = fmaxf(sh[threadIdx.x], sh[threadIdx.x + o]);
        __syncthreads();
    }
    if (threadIdx.x == 0) dst[0] = sh[0];
}

__global__ __launch_bounds__(256)
void k_scale_inv(float* __restrict__ dst, const float* __restrict__ src,
                 const float* __restrict__ slot) {
    const float s = 1.f / (slot[0] + EPSV);
    const int e = blockIdx.x * 256 + threadIdx.x;
    dst[e]      = s * src[e];
    dst[NN + e] = s * src[NN + e];
}

__global__ __launch_bounds__(256)
void k_scale_invtrace(float* __restrict__ x, const float* __restrict__ sc) {
    const float tr = sc[0], ti = sc[1];
    const float den = tr * tr + ti * ti + 1e-30f;
    const float cr = tr / den, ci = -ti / den;   // 1/trace
    const int e = blockIdx.x * 256 + threadIdx.x;
    const float xr = x[e], xi = x[NN + e];
    x[e]      = cr * xr - ci * xi;
    x[NN + e] = cr * xi + ci * xr;
}

// Sherman-Morrison rank-1 correction: w -= (t/d) * Mu, t=SC[0:2], d=SC[4:6]
__global__ __launch_bounds__(256)
void k_sm_update(float* __restrict__ w, const float* __restrict__ Mu,
                 const float* __restrict__ sc) {
    const float tr = sc[0], ti = sc[1], dr = sc[4], di = sc[5];
    const float den = dr * dr + di * di + 1e-30f;
    const float cr = (tr * dr + ti * di) / den;
    const float ci = (ti * dr - tr * di) / den;
    const int e = blockIdx.x * 256 + threadIdx.x;
    const float mr = Mu[e], mi = Mu[NN + e];
    w[e]      -= cr * mr - ci * mi;
    w[NN + e] -= cr * mi + ci * mr;
}

// h[k] = <V_k, w> for k <= j (else 0) -> Hs column j, rows 0..32
__global__ __launch_bounds__(256)
void k_dots33(const float* __restrict__ Vb, const float* __restrict__ w,
              float* __restrict__ Hs, int j) {
    const int k = blockIdx.x;
    __shared__ float sr[256], si[256];
    float ar = 0.f, ai = 0.f;
    if (k <= j) {
        const float* vk = Vb + (size_t)k * CV;
        for (int e = threadIdx.x; e < NN; e += 256) {
            const float vr = vk[e], vi = vk[NN + e];
            const float wr = w[e], wi = w[NN + e];
            ar += vr * wr + vi * wi;       // conj(v) * w
            ai += vr * wi - vi * wr;
        }
    }
    sr[threadIdx.x] = ar; si[threadIdx.x] = ai;
    __syncthreads();
    for (int o = 128; o > 0; o >>= 1) {
        if (threadIdx.x < o) { sr[threadIdx.x] += sr[threadIdx.x + o]; si[threadIdx.x] += si[threadIdx.x + o]; }
        __syncthreads();
    }
    if (threadIdx.x == 0) {
        Hs[(k * KRY + j) * 2]     = sr[0];
        Hs[(k * KRY + j) * 2 + 1] = si[0];
    }
}

__global__ __launch_bounds__(256)
void k_whupdate(float* __restrict__ w, const float* __restrict__ Vb,
                const float* __restrict__ Hs, int j) {
    const int e = blockIdx.x * 256 + threadIdx.x;
    float accr = 0.f, acci = 0.f;
    for (int k = 0; k <= j; ++k) {
        const float hr = Hs[(k * KRY + j) * 2], hi = Hs[(k * KRY + j) * 2 + 1];
        const float* vk = Vb + (size_t)k * CV;
        const float vr = vk[e], vi = vk[NN + e];
        accr += hr * vr - hi * vi;
        acci += hr * vi + hi * vr;
    }
    w[e]      -= accr;
    w[NN + e] -= acci;
}

__global__ __launch_bounds__(1)
void k_set_hs(float* __restrict__ Hs, const float* __restrict__ slot, int j) {
    Hs[((j + 1) * KRY + j) * 2]     = slot[0];
    Hs[((j + 1) * KRY + j) * 2 + 1] = 0.f;
}

__global__ __launch_bounds__(1)
void k_copy_slot(float* __restrict__ dst, const float* __restrict__ src) {
    dst[0] = src[0];
    dst[1] = src[1];
}

__global__ __launch_bounds__(256)
void k_xupdate(float* __restrict__ x, const float* __restrict__ Vb,
               const float* __restrict__ Y) {
    const int e = blockIdx.x * 256 + threadIdx.x;
    float accr = 0.f, acci = 0.f;
    for (int k = 0; k < KRY; ++k) {
        const float yr = Y[2 * k], yi = Y[2 * k + 1];
        const float* vk = Vb + (size_t)k * CV;
        const float vr = vk[e], vi = vk[NN + e];
        accr += yr * vr - yi * vi;
        acci += yr * vi + yi * vr;
    }
    x[e]      += accr;
    x[NN + e] += acci;
}

__global__ __launch_bounds__(256)
void k_hermitize(float* __restrict__ dst, const float* __restrict__ x) {
    const int e = blockIdx.x * 256 + threadIdx.x;
    const int i = e >> 7, jj = e & 127;
    const int et = jj * NH + i;
    dst[e]      = 0.5f * (x[e] + x[et]);
    dst[NN + e] = 0.5f * (x[NN + e] - x[NN + et]);
}

// Normal-equations least squares (33x32 Hessenberg) in LDS, one block.
__global__ __launch_bounds__(256)
void k_leastsq(const float* __restrict__ Hs, const float* __restrict__ betaSlot,
               float* __restrict__ Yout) {
    __shared__ float Ar_[KRY * KRY], Ai_[KRY * KRY];
    __shared__ float br_[KRY], bi_[KRY], yr_[KRY], yi_[KRY];
    __shared__ float pvr, pvi;
    const int tid = threadIdx.x;
    for (int cidx = tid; cidx < KRY * KRY; cidx += 256) {
        const int i = cidx / KRY, jc = cidx % KRY;
        float sr = 0.f, si = 0.f;
        for (int r = 0; r < KRY + 1; ++r) {
            const float h1r = Hs[(r * KRY + i) * 2],  h1i = Hs[(r * KRY + i) * 2 + 1];
            const float h2r = Hs[(r * KRY + jc) * 2], h2i = Hs[(r * KRY + jc) * 2 + 1];
            sr += h1r * h2r + h1i * h2i;       // conj(H[:,i]) . H[:,jc]
            si += h1r * h2i - h1i * h2r;
        }
        if (i == jc) sr += 1e-6f;
        Ar_[cidx] = sr; Ai_[cidx] = si;
    }
    if (tid < KRY) {
        const float b = betaSlot[0];
        br_[tid] =  Hs[tid * 2] * b;           // conj(Hs[0,tid]) * beta
        bi_[tid] = -Hs[tid * 2 + 1] * b;
    }
    __syncthreads();
    for (int p = 0; p < KRY; ++p) {
        if (tid == 0) {
            const float dr = Ar_[p * KRY + p], di = Ai_[p * KRY + p];
            const float den = dr * dr + di * di + 1e-30f;
            pvr = dr / den; pvi = -di / den;
        }
        __syncthreads();
        if (tid > p && tid < KRY) {
            const float fr = Ar_[tid * KRY + p] * pvr - Ai_[tid * KRY + p] * pvi;
            const float fi = Ar_[tid * KRY + p] * pvi + Ai_[tid * KRY + p] * pvr;
            for (int jc = p; jc < KRY; ++jc) {
                const float ur = Ar_[p * KRY + jc], ui = Ai_[p * KRY + jc];
                Ar_[tid * KRY + jc] -= fr * ur - fi * ui;
                Ai_[tid * KRY + jc] -= fr * ui + fi * ur;
            }
            br_[tid] -= fr * br_[p] - fi * bi_[p];
            bi_[tid] -= fr * bi_[p] + fi * br_[p];
        }
        __syncthreads();
    }
    if (tid == 0) {
        for (int i = KRY - 1; i >= 0; --i) {
            float sr = br_[i], si = bi_[i];
            for (int jc = i + 1; jc < KRY; ++jc) {
                sr -= Ar_[i * KRY + jc] * yr_[jc] - Ai_[i * KRY + jc] * yi_[jc];
                si -= Ar_[i * KRY + jc] * yi_[jc] + Ai_[i * KRY + jc] * yr_[jc];
            }
            const float dr = Ar_[i * KRY + i], di = Ai_[i * KRY + i];
            const float den = dr * dr + di * di + 1e-30f;
            yr_[i] = (sr * dr + si * di) / den;
            yi_[i] = (si * dr - sr * di) / den;
        }
    }
    __syncthreads();
    if (tid < KRY) { Yout[2 * tid] = yr_[tid]; Yout[2 * tid + 1] = yi_[tid]; }
}

__global__ __launch_bounds__(256)
void k_writeout(float* __restrict__ out, const float* __restrict__ rho, int out_size) {
    const int e = blockIdx.x * 256 + threadIdx.x;
    if (2 * e < out_size)     out[2 * e]     = rho[e];
    if (2 * e + 1 < out_size) out[2 * e + 1] = rho[NN + e];
}

__global__ __launch_bounds__(1)
void k_finalize(const float* __restrict__ res, float* __restrict__ out, int out_size) {
    int nit = MAXIT, succ = 0;
    for (int r = 0; r < MAXIT; ++r) {
        if (res[r] < TOLV) { nit = r + 1; succ = 1; break; }
    }
    if (2 * NN < out_size)     out[2 * NN]     = (float)nit;
    if (2 * NN + 1 < out_size) out[2 * NN + 1] = (float)succ;
}

// ---------------------------------------------------------------------------
// Host orchestration (all launches on `stream`, graph-capture safe).
// ---------------------------------------------------------------------------
struct CM { const float* r; const float* i; };

struct Ctx {
    hipStream_t s;
    CM H, L0, L1;
    float *LdagL, *G, *X, *AW, *W, *RR, *TT, *MU, *XM, *VB, *Hs, *Y, *SC, *RES;
};

static inline CM cmv(const float* base) { CM m; m.r = base; m.i = base + NN; return m; }

static void gemm(const Ctx& c, CM A, int opA, CM B, int opB, float* Cb,
                 float alr, float ali, float beta) {
    czgemm128<<<dim3(16), dim3(128), 0, c.s>>>(A.r, A.i, opA, B.r, B.i, opB,
                                               Cb, Cb + NN, alr, ali, beta);
}

// out = lindblad(rho): -i(H rho - rho H) + sum_k L rho L^dag - 0.5(LL rho + rho LL)
static void lindblad(const Ctx& c, const float* rho, float* out) {
    const CM R = cmv(rho);
    gemm(c, c.H, 0, R, 0, out, 0.f, -1.f, 0.f);           // -i H rho
    gemm(c, R, 0, c.H, 0, out, 0.f,  1.f, 1.f);           // +i rho H
    gemm(c, c.L0, 0, R, 0, c.TT, 1.f, 0.f, 0.f);          // T = L0 rho
    gemm(c, cmv(c.TT), 0, c.L0, 1, out, 1.f, 0.f, 1.f);   // += T L0^dag
    gemm(c, c.L1, 0, R, 0, c.TT, 1.f, 0.f, 0.f);          // T = L1 rho
    gemm(c, cmv(c.TT), 0, c.L1, 1, out, 1.f, 0.f, 1.f);   // += T L1^dag
    gemm(c, cmv(c.LdagL), 0, R, 0, out, -0.5f, 0.f, 1.f); // -0.5 LL rho
    gemm(c, R, 0, cmv(c.LdagL), 0, out, -0.5f, 0.f, 1.f); // -0.5 rho LL
}

// A_op(x) = vec(lindblad(x)) + trace(x) * vec(I)
static void a_op(const Ctx& c, const float* x, float* out) {
    lindblad(c, x, out);
    k_trace<<<1, 128, 0, c.s>>>(x, c.SC, 0);
    k_add_diag<<<1, 128, 0, c.s>>>(out, c.SC);
}

// base_prec(Y) = -X where X approx solves G X + X G^dag = Y (Richardson, tau=SC[2])
static void base_prec(const Ctx& c, const float* Y, float* X) {
    k_scale_dev<<<64, 256, 0, c.s>>>(X, Y, c.SC + 2);
    for (int it = 0; it < LYAP_IT; ++it) {
        k_copy<<<64, 256, 0, c.s>>>(c.RR, Y);
        gemm(c, cmv(c.G), 0, cmv(X), 0, c.RR, -1.f, 0.f, 1.f);
        gemm(c, cmv(X), 0, cmv(c.G), 1, c.RR, -1.f, 0.f, 1.f);
        k_axpy_dev<<<64, 256, 0, c.s>>>(X, c.RR, c.SC + 2);
    }
    k_scale_host<<<64, 256, 0, c.s>>>(X, -1.f, 0.f);
}

// full preconditioner with Sherman-Morrison rank-1 update
static void precond(const Ctx& c, const float* Y, float* X) {
    base_prec(c, Y, X);
    k_trace<<<1, 128, 0, c.s>>>(X, c.SC, 0);              // <u, mx> = trace(mx)
    k_sm_update<<<64, 256, 0, c.s>>>(X, c.MU, c.SC);
}

extern "C" void kernel_launch(void* const* d_in, const int* in_sizes, int n_in,
                              void* d_out, int out_size, void* d_ws, size_t ws_size,
                              hipStream_t stream) {
    const float* Hr  = (const float*)d_in[0];
    const float* Hi  = (const float*)d_in[1];
    const float* Lsr = (const float*)d_in[2];
    const float* Lsi = (const float*)d_in[3];
    float* ws = (float*)d_ws;

    const size_t need = (size_t)(9 + (KRY + 1)) * CV + (KRY + 1) * KRY * 2 + KRY * 2 + 32 + MAXIT;
    if (ws_size < need * sizeof(float)) return;

    Ctx c;
    c.s = stream;
    c.H.r  = Hr;  c.H.i  = Hi;
    c.L0.r = Lsr; c.L0.i = Lsi;
    c.L1.r = Lsr + NN; c.L1.i = Lsi + NN;
    size_t o = 0;
    c.LdagL = ws + o; o += CV;
    c.G  = ws + o; o += CV;
    c.X  = ws + o; o += CV;
    c.AW = ws + o; o += CV;
    c.W  = ws + o; o += CV;
    c.RR = ws + o; o += CV;
    c.TT = ws + o; o += CV;
    c.MU = ws + o; o += CV;
    c.XM = ws + o; o += CV;
    c.VB = ws + o; o += (size_t)(KRY + 1) * CV;
    c.Hs = ws + o; o += (KRY + 1) * KRY * 2;
    c.Y  = ws + o; o += KRY * 2;
    c.SC = ws + o; o += 32;
    c.RES = ws + o; o += MAXIT;

    // ---- setup -----------------------------------------------------------
    hipMemsetAsync(c.X, 0, CV * sizeof(float), stream);
    hipMemsetAsync(c.SC, 0, 32 * sizeof(float), stream);
    k_set_entry<<<1, 1, 0, stream>>>(c.X);                // x0 = e_0 (rho0=|0><0|)
    gemm(c, c.L0, 1, c.L0, 0, c.LdagL, 1.f, 0.f, 0.f);    // L0^dag L0
    gemm(c, c.L1, 1, c.L1, 0, c.LdagL, 1.f, 0.f, 1.f);    // + L1^dag L1
    k_buildG<<<64, 256, 0, stream>>>(c.G, Hr, Hi, c.LdagL);
    k_fro_tau<<<1, 256, 0, stream>>>(c.G, c.SC + 2);      // tau = 0.5/||G||_F
    k_set_identity<<<64, 256, 0, stream>>>(c.AW);         // u = vec(I)
    base_prec(c, c.AW, c.MU);                             // Mu = base_prec(u)
    k_trace<<<1, 128, 0, stream>>>(c.MU, c.SC + 4, 1);    // smden = 1 + <u,Mu>

    // ---- GMRES restarts --------------------------------------------------
    for (int r = 0; r < MAXIT; ++r) {
        a_op(c, c.X, c.AW);
        k_neg_plus_id<<<64, 256, 0, stream>>>(c.AW);      // r = u - A x
        precond(c, c.AW, c.W);                            // r_p = P(r)
        k_norm<<<1, 256, 0, stream>>>(c.W, c.SC + 3);     // beta
        k_copy_slot<<<1, 1, 0, stream>>>(c.SC + 6, c.SC + 3);
        k_scale_inv<<<64, 256, 0, stream>>>(c.VB, c.W, c.SC + 3);   // V0

        for (int j = 0; j < KRY; ++j) {
            a_op(c, c.VB + (size_t)j * CV, c.AW);
            precond(c, c.AW, c.W);                        // w = P(A v_j)
            k_dots33<<<KRY + 1, 256, 0, stream>>>(c.VB, c.W, c.Hs, j);
            k_whupdate<<<64, 256, 0, stream>>>(c.W, c.VB, c.Hs, j);
            k_norm<<<1, 256, 0, stream>>>(c.W, c.SC + 3);
            k_set_hs<<<1, 1, 0, stream>>>(c.Hs, c.SC + 3, j);
            k_scale_inv<<<64, 256, 0, stream>>>(c.VB + (size_t)(j + 1) * CV, c.W, c.SC + 3);
        }

        k_leastsq<<<1, 256, 0, stream>>>(c.Hs, c.SC + 6, c.Y);
        k_xupdate<<<64, 256, 0, stream>>>(c.X, c.VB, c.Y);

        // stopping-criterion residual on hermitized, trace-normalized x
        k_hermitize<<<64, 256, 0, stream>>>(c.XM, c.X);
        k_trace<<<1, 128, 0, stream>>>(c.XM, c.SC, 0);
        k_scale_invtrace<<<64, 256, 0, stream>>>(c.XM, c.SC);
        lindblad(c, c.XM, c.AW);
        k_maxabs<<<1, 256, 0, stream>>>(c.AW, c.RES + r);
    }

    // rho = hermitize(x)/trace already in XM from the last restart
    k_writeout<<<64, 256, 0, stream>>>((float*)d_out, c.XM, out_size);
    k_finalize<<<1, 1, 0, stream>>>(c.RES, (float*)d_out, out_size);
}